// ConditionalFlowMatchingOTLoss_22471268892863
// MI455X (gfx1250) — compile-verified
//
#include <hip/hip_runtime.h>
#include <hip/hip_bf16.h>

#define BATCH   512
#define NPTS    150
#define NPAD    160
#define FDIM    3
#define HDIM    512
#define MSTRIDE 151        // odd stride -> conflict-free column walks in 64-bank LDS
#define OT_EPSF 0.01f
#define OT_IEPS 100.0f
#define OT_ITER 100
#define SIGMAF  0.0001f

typedef __attribute__((ext_vector_type(2))) float v2f;
typedef __attribute__((ext_vector_type(8))) float v8f;

// ---------------- LDS layout (floats) ----------------
// M:22650  X0:450 X1:450 F:152 G:152 RED:256 YT:640 UT:480 MOT:160 HS:8*16*17
#define OFF_M   0
#define OFF_X0  22650
#define OFF_X1  (OFF_X0 + 450)
#define OFF_F   (OFF_X1 + 450)
#define OFF_G   (OFF_F + 152)
#define OFF_RED (OFF_G + 152)
#define OFF_YT  (OFF_RED + 256)
#define OFF_UT  (OFF_YT + 640)
#define OFF_MOT (OFF_UT + 480)
#define OFF_HS  (OFF_MOT + 160)
#define SMEM_FLOATS (OFF_HS + 8*16*17)

// ---------------- JAX threefry2x32 ----------------
__device__ __forceinline__ unsigned rotl32(unsigned x, int r){ return (x<<r)|(x>>(32-r)); }

__device__ inline void tf2x32(unsigned k0, unsigned k1, unsigned c0, unsigned c1,
                              unsigned& o0, unsigned& o1){
  const unsigned ks2 = k0 ^ k1 ^ 0x1BD11BDAu;
  unsigned x0 = c0 + k0, x1 = c1 + k1;
  x0+=x1; x1=rotl32(x1,13); x1^=x0;  x0+=x1; x1=rotl32(x1,15); x1^=x0;
  x0+=x1; x1=rotl32(x1,26); x1^=x0;  x0+=x1; x1=rotl32(x1, 6); x1^=x0;
  x0+=k1;  x1+=ks2+1u;
  x0+=x1; x1=rotl32(x1,17); x1^=x0;  x0+=x1; x1=rotl32(x1,29); x1^=x0;
  x0+=x1; x1=rotl32(x1,16); x1^=x0;  x0+=x1; x1=rotl32(x1,24); x1^=x0;
  x0+=ks2; x1+=k0+2u;
  x0+=x1; x1=rotl32(x1,13); x1^=x0;  x0+=x1; x1=rotl32(x1,15); x1^=x0;
  x0+=x1; x1=rotl32(x1,26); x1^=x0;  x0+=x1; x1=rotl32(x1, 6); x1^=x0;
  x0+=k0;  x1+=k1+3u;
  x0+=x1; x1=rotl32(x1,17); x1^=x0;  x0+=x1; x1=rotl32(x1,29); x1^=x0;
  x0+=x1; x1=rotl32(x1,16); x1^=x0;  x0+=x1; x1=rotl32(x1,24); x1^=x0;
  x0+=k1;  x1+=ks2+4u;
  x0+=x1; x1=rotl32(x1,13); x1^=x0;  x0+=x1; x1=rotl32(x1,15); x1^=x0;
  x0+=x1; x1=rotl32(x1,26); x1^=x0;  x0+=x1; x1=rotl32(x1, 6); x1^=x0;
  x0+=ks2; x1+=k0+5u;
  o0 = x0; o1 = x1;
}

// JAX random_bits element `idx` of a flat draw of `total` (even) uint32s
__device__ inline unsigned jax_bits(unsigned k0, unsigned k1,
                                    unsigned long long idx, unsigned long long total){
  unsigned long long hf = total >> 1;
  unsigned o0, o1;
  if (idx < hf){ tf2x32(k0,k1,(unsigned)idx,(unsigned)(idx+hf),o0,o1); return o0; }
  tf2x32(k0,k1,(unsigned)(idx-hf),(unsigned)idx,o0,o1); return o1;
}

// branch-free tanh: exact in f32 for |z|>=15 anyway; keeps EXEC untouched
__device__ __forceinline__ float tanh_fast(float z){
  float zc = fminf(fmaxf(z, -15.0f), 15.0f);
  float e  = __expf(2.0f * zc);          // single v_exp_f32
  return 1.0f - __fdividef(2.0f, e + 1.0f);
}

__device__ __forceinline__ float gelu_f(float v){
  float v3 = v*v*v;
  return 0.5f*v*(1.0f + tanh_fast(0.7978845608028654f*(v + 0.044715f*v3)));
}

// ================= mega-kernel: one workgroup per batch sample =================
__global__ __launch_bounds__(256)
void cfm_ot_kernel(const float* __restrict__ x,    const float* __restrict__ mask,
                   const float* __restrict__ t,    const float* __restrict__ x0n,
                   const float* __restrict__ epsn, const float* __restrict__ W1,
                   const float* __restrict__ b1,   const float* __restrict__ W2,
                   const float* __restrict__ b2,   float* __restrict__ ws)
{
  extern __shared__ float smem[];
  float* Mm  = smem + OFF_M;
  float* X0  = smem + OFF_X0;
  float* X1  = smem + OFF_X1;
  float* Fv  = smem + OFF_F;
  float* Gv  = smem + OFF_G;
  float* RED = smem + OFF_RED;
  float* YT  = smem + OFF_YT;   // [160][4]  MLP input rows [y0 y1 y2 t]
  float* UT  = smem + OFF_UT;   // [160][3]
  float* MOT = smem + OFF_MOT;  // [160]

  const int b   = blockIdx.x;
  const int tid = threadIdx.x;

  // warm L2 with the shared MLP weights (global_prefetch_b8 path)
  if (tid < 8){
    __builtin_prefetch(W1 + tid*256, 0, 1);          // 4x512 f32 = 8KB
    __builtin_prefetch(W2 + tid*192, 0, 1);          // 512x3 f32 = 6KB
    __builtin_prefetch(b1 + tid*64, 0, 1);           // 512 f32   = 2KB
  }

  // ---- stage point clouds ----
  for (int k = tid; k < NPTS*FDIM; k += 256){
    X0[k] = x0n[b*NPTS*FDIM + k];   // x0 = x0_noise
    X1[k] = x  [b*NPTS*FDIM + k];   // x1 = x
  }
  __syncthreads();

  // ---- cost matrix + per-batch max ----
  float lmax = 0.0f;
  for (int e = tid; e < NPTS*NPTS; e += 256){
    int i = e / NPTS, j = e - i*NPTS;
    float d0 = X0[i*3+0]-X1[j*3+0];
    float d1 = X0[i*3+1]-X1[j*3+1];
    float d2 = X0[i*3+2]-X1[j*3+2];
    float m  = d0*d0 + d1*d1 + d2*d2;
    Mm[i*MSTRIDE+j] = m;
    lmax = fmaxf(lmax, m);
  }
  RED[tid] = lmax;
  __syncthreads();
  for (int s = 128; s > 0; s >>= 1){
    if (tid < s) RED[tid] = fmaxf(RED[tid], RED[tid+s]);
    __syncthreads();
  }
  float inv_max = 1.0f / RED[0];
  for (int e = tid; e < NPTS*NPTS; e += 256){
    int i = e / NPTS, j = e - i*NPTS;
    Mm[i*MSTRIDE+j] *= inv_max;
  }
  if (tid < NPTS){ Fv[tid] = 0.0f; Gv[tid] = 0.0f; }
  __syncthreads();

  // ---- 100 log-domain Sinkhorn iterations, fully in LDS ----
  const float eps_loga = OT_EPSF * (-5.0106352940962555f); // eps * (-log N)
  for (int it = 0; it < OT_ITER; ++it){
    if (tid < NPTS){                       // g[j]: reduce over i (stride MSTRIDE, odd)
      const int j = tid;
      float mx = -3.4e38f;
      for (int i = 0; i < NPTS; ++i) mx = fmaxf(mx, Fv[i] - Mm[i*MSTRIDE+j]);
      float s = 0.0f;
      for (int i = 0; i < NPTS; ++i) s += __expf((Fv[i] - Mm[i*MSTRIDE+j] - mx)*OT_IEPS);
      Gv[j] = eps_loga - mx - OT_EPSF*__logf(s);
    }
    __syncthreads();
    if (tid < NPTS){                       // f[i]: reduce over j (unit stride)
      const int i = tid;
      float mx = -3.4e38f;
      for (int j = 0; j < NPTS; ++j) mx = fmaxf(mx, Gv[j] - Mm[i*MSTRIDE+j]);
      float s = 0.0f;
      for (int j = 0; j < NPTS; ++j) s += __expf((Gv[j] - Mm[i*MSTRIDE+j] - mx)*OT_IEPS);
      Fv[i] = eps_loga - mx - OT_EPSF*__logf(s);
    }
    __syncthreads();
  }

  // ---- threefry keys: key(42) split -> ki, kj ----
  unsigned ki0, ki1, kj0, kj1;
  { unsigned a0,a1,c0,c1;
    tf2x32(0u, 42u, 0u, 2u, a0, c0);   // split lane 0
    tf2x32(0u, 42u, 1u, 3u, a1, c1);   // split lane 1
    ki0=a0; ki1=a1; kj0=c0; kj1=c1;
  }
  const float tb = t[b];

  // ---- per-row sampling: i ~ randint, j ~ gumbel-argmax over logpi row ----
  if (tid < NPTS){
    const int n = tid;
    unsigned bits = jax_bits(ki0, ki1,
                             (unsigned long long)(b*NPTS + n),
                             (unsigned long long)BATCH*NPTS);
    const int isel = (int)(bits % (unsigned)NPTS);
    const float fi = Fv[isel];
    const unsigned long long base = ((unsigned long long)(b*NPTS + n)) * NPTS;
    const unsigned long long totg = (unsigned long long)BATCH*NPTS*NPTS;
    float best = -3.4e38f; int jsel = 0;
    for (int j = 0; j < NPTS; ++j){
      unsigned gb = jax_bits(kj0, kj1, base + j, totg);
      float u = __uint_as_float(0x3f800000u | (gb >> 9)) - 1.0f;
      u = u * (1.0f - 1.1754944e-38f) + 1.1754944e-38f;
      float gum = -__logf(-__logf(u));
      float sc  = (fi + Gv[j] - Mm[isel*MSTRIDE+j]) * OT_IEPS + gum;
      if (sc > best){ best = sc; jsel = j; }
    }
    const float mot = mask[b*NPTS + jsel];
    MOT[n] = mot;
    #pragma unroll
    for (int c = 0; c < 3; ++c){
      float x0v = X0[isel*3+c], x1v = X1[jsel*3+c];
      float mu  = x0v*tb + x1v*(1.0f - tb);
      YT[n*4+c] = mu + SIGMAF * epsn[(b*NPTS+n)*3 + c];
      UT[n*3+c] = (x0v - x1v) * mot;
    }
    YT[n*4+3] = tb;
  } else if (tid < NPAD){
    const int n = tid;                    // zero pad rows 150..159
    MOT[n] = 0.0f;
    YT[n*4+0]=0.0f; YT[n*4+1]=0.0f; YT[n*4+2]=0.0f; YT[n*4+3]=0.0f;
    UT[n*3+0]=0.0f; UT[n*3+1]=0.0f; UT[n*3+2]=0.0f;
  }
  __syncthreads();

  // ---- MLP via V_WMMA_F32_16X16X4_F32: (160x4 @ 4x512) -> gelu -> (@ 512x3) ----
  const int  wave   = tid >> 5;
  const int  lane   = tid & 31;
  const int  lo     = lane & 15;
  const bool hiHalf = lane >= 16;
  float* hs = smem + OFF_HS + wave*(16*17);   // wave-private 16x16 staging (pad 17)

  float lsum = 0.0f;
  for (int tile = wave; tile < 10; tile += 8){   // 10 row-tiles of 16
    const int row = tile*16 + lo;
    v2f afrag;                                   // A 16x4: lanes 0-15 K=0,1; 16-31 K=2,3
    afrag.x = YT[row*4 + (hiHalf ? 2 : 0)];
    afrag.y = YT[row*4 + (hiHalf ? 3 : 1)];
    v8f acc2 = {0.f,0.f,0.f,0.f,0.f,0.f,0.f,0.f};

    for (int nt = 0; nt < 32; ++nt){             // 32 column tiles of H=512
      const int ncol = nt*16 + lo;
      const int kb   = hiHalf ? 2 : 0;
      v2f bfrag;                                 // B 4x16: lanes 0-15 K=0,1; 16-31 K=2,3
      bfrag.x = W1[ kb   *HDIM + ncol];
      bfrag.y = W1[(kb+1)*HDIM + ncol];
      v8f c = {0.f,0.f,0.f,0.f,0.f,0.f,0.f,0.f};
      c = __builtin_amdgcn_wmma_f32_16x16x4_f32(false, afrag, false, bfrag,
                                                (short)0, c, false, false);
      // bias + gelu (branch-free); C layout: vgpr v, lane -> (M = v + 8*hi, N = lo)
      const float bias = b1[ncol];
      #pragma unroll
      for (int v = 0; v < 8; ++v){
        int m = v + (hiHalf ? 8 : 0);
        hs[m*17 + lo] = gelu_f(c[v] + bias);
      }
      // GEMM2: accumulate 4 K=4 chunks of this h tile against W2 (cols 0..2 valid)
      #pragma unroll
      for (int kk = 0; kk < 4; ++kk){
        const int klocal = kk*4 + (hiHalf ? 2 : 0);
        v2f a2;
        a2.x = hs[lo*17 + klocal];
        a2.y = hs[lo*17 + klocal + 1];
        const int kg = nt*16 + klocal;
        v2f b2f;
        b2f.x = (lo < FDIM) ? W2[ kg   *FDIM + lo] : 0.0f;
        b2f.y = (lo < FDIM) ? W2[(kg+1)*FDIM + lo] : 0.0f;
        acc2 = __builtin_amdgcn_wmma_f32_16x16x4_f32(false, a2, false, b2f,
                                                     (short)0, acc2, false, false);
      }
    }
    // epilogue: vt = (acc2 + b2) * mask_ot ; accumulate squared error
    if (lo < FDIM){
      const float bias2 = b2[lo];
      #pragma unroll
      for (int v = 0; v < 8; ++v){
        int m = tile*16 + v + (hiHalf ? 8 : 0);
        if (m < NPTS){
          float vt = (acc2[v] + bias2) * MOT[m];
          float d  = vt - UT[m*3 + lo];
          lsum += d*d;
        }
      }
    }
  }

  // ---- block loss reduction -> ws[b] ----
  __syncthreads();
  RED[tid] = lsum;
  __syncthreads();
  for (int s = 128; s > 0; s >>= 1){
    if (tid < s) RED[tid] += RED[tid+s];
    __syncthreads();
  }
  if (tid == 0) ws[b] = RED[0];
}

// ================= final mean over 512 block sums =================
__global__ __launch_bounds__(256)
void cfm_finalize_kernel(const float* __restrict__ ws, float* __restrict__ out){
  __shared__ float red[256];
  const int tid = threadIdx.x;
  red[tid] = ws[tid] + ws[tid + 256];
  __syncthreads();
  for (int s = 128; s > 0; s >>= 1){
    if (tid < s) red[tid] += red[tid+s];
    __syncthreads();
  }
  if (tid == 0) out[0] = red[0] / (float)(BATCH * NPTS * FDIM);
}

extern "C" void kernel_launch(void* const* d_in, const int* in_sizes, int n_in,
                              void* d_out, int out_size, void* d_ws, size_t ws_size,
                              hipStream_t stream){
  const float* x    = (const float*)d_in[0];
  const float* mask = (const float*)d_in[1];
  const float* t    = (const float*)d_in[2];
  const float* x0n  = (const float*)d_in[3];
  const float* epsn = (const float*)d_in[4];
  const float* W1   = (const float*)d_in[5];
  const float* b1   = (const float*)d_in[6];
  const float* W2   = (const float*)d_in[7];
  const float* b2   = (const float*)d_in[8];
  float* ws  = (float*)d_ws;
  float* out = (float*)d_out;

  const size_t smem_bytes = (size_t)SMEM_FLOATS * sizeof(float); // ~110 KB < 320 KB/WGP
  cfm_ot_kernel<<<BATCH, 256, smem_bytes, stream>>>(x, mask, t, x0n, epsn,
                                                    W1, b1, W2, b2, ws);
  cfm_finalize_kernel<<<1, 256, 0, stream>>>(ws, out);
}